// QuantumGate_16862041604144
// MI455X (gfx1250) — compile-verified
//
#include <hip/hip_runtime.h>
#include <hip/hip_bf16.h>

// V_WMMA_F32_16X16X4_F32:  D(16x16) = A(16x4) * B(4x16) + C, all f32.
typedef float v2f __attribute__((ext_vector_type(2)));
typedef float v8f __attribute__((ext_vector_type(8)));

#define TOTAL_QUBITS 24
#define BATCH 4
#define ITERS 8            // 8 iterations * 32 cols = 256 base indices per wave
#define WAVES_PER_BLOCK 8  // 256 threads

// Insert a zero bit at final-domain position pA, then pB (pA < pB).
__device__ __forceinline__ unsigned expand2(unsigned x, unsigned maskA, unsigned maskB) {
    x = ((x & ~maskA) << 1) | (x & maskA);
    x = ((x & ~maskB) << 1) | (x & maskB);
    return x;
}

__global__ __launch_bounds__(256) void quantum_gate2_wmma(
    const float* __restrict__ state,   // [BATCH, 2^24]
    const float* __restrict__ mat,     // [4,4] row-major
    const int*   __restrict__ q0p,
    const int*   __restrict__ q1p,
    float*       __restrict__ out)     // [BATCH, 2^24]
{
    const unsigned lane = threadIdx.x & 31u;
    const unsigned warp = threadIdx.x >> 5;

    const int q0 = *q0p;
    const int q1 = *q1p;
    const int b0pos = (TOTAL_QUBITS - 1) - q0;     // bit of q0 in flat index (k-bit 1)
    const int b1pos = (TOTAL_QUBITS - 1) - q1;     // bit of q1 in flat index (k-bit 0)
    const unsigned m_q0  = 1u << b0pos;
    const unsigned m_q1  = 1u << b1pos;
    const unsigned m_q01 = m_q0 + m_q1;
    const int pA = b0pos < b1pos ? b0pos : b1pos;
    const int pB = b0pos < b1pos ? b1pos : b0pos;
    const unsigned maskA = (1u << pA) - 1u;
    const unsigned maskB = (1u << pB) - 1u;

    // Scalar batch decomposition (blockIdx-derived => SADDR-form memory ops).
    // 2^22 free indices per batch / 256 per wave / 8 waves per block = 2048 blocks/batch.
    const unsigned blocks_per_batch = 1u << (TOTAL_QUBITS - 2 - 8 - 3);
    const unsigned batch  = blockIdx.x / blocks_per_batch;
    const unsigned blkib  = blockIdx.x & (blocks_per_batch - 1u);
    const unsigned free_base = ((blkib << 3) + warp) << 8;           // 256 cols/wave

    const size_t boff = (size_t)batch << TOTAL_QUBITS;
    const float* __restrict__ sb = state + boff;
    float*       __restrict__ ob = out   + boff;

    // Lane roles. A layout (16x4 f32): V0 = {lanes0-15: K=0 | lanes16-31: K=2},
    // V1 = {K=1 | K=3}, row M = lane%16.  B layout symmetric with N = lane%16.
    const unsigned r    = lane & 15u;
    const unsigned kA   = (lane >> 4) << 1;            // 0 or 2
    const unsigned koff = (lane >= 16u) ? m_q0 : 0u;   // lane half -> k rows {0,1} vs {2,3}

    // A1: gate in rows 0..3  (produces D rows 0..3  -> lanes 0..15 of v0..v3)
    // A2: gate in rows 8..11 (produces D rows 8..11 -> lanes 16..31 of v0..v3)
    const bool in1 = (r < 4u);
    const bool in2 = (r >= 8u) && (r < 12u);
    v2f A1, A2;
    A1.x = in1 ? mat[r * 4u + kA]              : 0.0f;
    A1.y = in1 ? mat[r * 4u + kA + 1u]         : 0.0f;
    A2.x = in2 ? mat[(r - 8u) * 4u + kA]       : 0.0f;
    A2.y = in2 ? mat[(r - 8u) * 4u + kA + 1u]  : 0.0f;

#pragma unroll
    for (int t = 0; t < ITERS; ++t) {
        const unsigned base = free_base + (unsigned)t * 32u;
        // Load-columns: B1 covers cols base+0..15, B2 covers cols base+16..31.
        const unsigned eL1 = expand2(base + r,       maskA, maskB);
        const unsigned eL2 = expand2(base + r + 16u, maskA, maskB);
        // Store column for this lane is exactly `lane` within the 32-block:
        const unsigned eS  = (lane < 16u) ? eL1 : eL2;

        // k = 2*bit(q0) + bit(q1): k0=e, k1=e+m_q1, k2=e+m_q0, k3=e+m_q01.
        const unsigned a1 = eL1 + koff;
        const unsigned a2 = eL2 + koff;
        v2f B1, B2;
        B1.x = sb[a1];             // k = 0 or 2
        B1.y = sb[a1 + m_q1];      // k = 1 or 3
        B2.x = sb[a2];
        B2.y = sb[a2 + m_q1];

        v8f C = {};
        v8f D = __builtin_amdgcn_wmma_f32_16x16x4_f32(
            false, A1, false, B1, (short)0, C, false, false);
        D = __builtin_amdgcn_wmma_f32_16x16x4_f32(
            false, A2, false, B2, (short)0, D, false, false);

        // v0..v3 = outputs g=0..3; lanes 0..15 cols 0..15 (rows 0..3),
        // lanes 16..31 cols 16..31 (rows 8..11): 32 contiguous floats per store.
        ob[eS]          = D[0];    // g=0
        ob[eS + m_q1]   = D[1];    // g=1
        ob[eS + m_q0]   = D[2];    // g=2
        ob[eS + m_q01]  = D[3];    // g=3
    }
}

extern "C" void kernel_launch(void* const* d_in, const int* in_sizes, int n_in,
                              void* d_out, int out_size, void* d_ws, size_t ws_size,
                              hipStream_t stream) {
    const float* state  = (const float*)d_in[0];
    const float* matrix = (const float*)d_in[1];
    const int*   q0     = (const int*)d_in[2];
    const int*   q1     = (const int*)d_in[3];
    float*       out    = (float*)d_out;

    // BATCH * 2048 blocks, 256 threads each.
    const unsigned blocks_per_batch = 1u << (TOTAL_QUBITS - 2 - 8 - 3); // 2048
    const unsigned blocks = (unsigned)BATCH * blocks_per_batch;         // 8192

    quantum_gate2_wmma<<<dim3(blocks), dim3(256), 0, stream>>>(state, matrix, q0, q1, out);
}